// ConvCRF_549755813891
// MI455X (gfx1250) — compile-verified
//
#include <hip/hip_runtime.h>

#define BATCH 4
#define H 512
#define W 512
#define C 21
#define HB 128
#define WB 128
#define SPAN 3

typedef __attribute__((ext_vector_type(2))) float v2f;
typedef __attribute__((ext_vector_type(8))) float v8f;

// ---------------------------------------------------------------------------
// Kernel 1: 4x4 average-pool of rgb, scaled by 1/THETA_ALPHA.
// Output layout: [b][y][x][3] on the 128x128 grid.
// ---------------------------------------------------------------------------
__global__ __launch_bounds__(256) void pool_rgb_kernel(
    const float* __restrict__ rgb, float* __restrict__ rgb_ds) {
  int idx = blockIdx.x * blockDim.x + threadIdx.x;
  if (idx >= BATCH * HB * WB) return;
  int x  = idx % WB;
  int y  = (idx / WB) % HB;
  int bb = idx / (WB * HB);

  const float* base = rgb + (size_t)bb * H * W * 3;
  float s0 = 0.f, s1 = 0.f, s2 = 0.f;
#pragma unroll
  for (int dy = 0; dy < 4; ++dy) {
    const float* row = base + ((size_t)(y * 4 + dy) * W + (size_t)x * 4) * 3;
#pragma unroll
    for (int dx = 0; dx < 4; ++dx) {
      s0 += row[dx * 3 + 0];
      s1 += row[dx * 3 + 1];
      s2 += row[dx * 3 + 2];
    }
  }
  const float sc = 1.0f / (16.0f * 160.0f);  // mean of 16, then /THETA_ALPHA
  float* o = rgb_ds + (size_t)idx * 3;
  o[0] = s0 * sc;
  o[1] = s1 * sc;
  o[2] = s2 * sc;
}

// ---------------------------------------------------------------------------
// Kernel 2: per low-res pixel, 49-tap Gaussian sums.
//   sn(y,x) = sum_{in-bounds offsets} exp(-0.5 * (16/9)(dx^2+dy^2))
//   bn(y,x) = sum exp(-0.5 * ((16/9)(dx^2+dy^2) + ||rgb_n - rgb_c||^2))
//   spf = sn/(sn+EPS); blf = sn/(bn+EPS)
// ---------------------------------------------------------------------------
__global__ __launch_bounds__(256) void fields_kernel(
    const float* __restrict__ rgb_ds,
    float* __restrict__ spf, float* __restrict__ blf) {
  int idx = blockIdx.x * blockDim.x + threadIdx.x;
  if (idx >= BATCH * HB * WB) return;
  int x  = idx % WB;
  int y  = (idx / WB) % HB;
  int bb = idx / (WB * HB);

  const float* base = rgb_ds + (size_t)bb * HB * WB * 3;
  const float* cpx  = base + ((size_t)y * WB + x) * 3;
  float r0 = cpx[0], r1 = cpx[1], r2 = cpx[2];

  float sn = 0.f, bn = 0.f;
#pragma unroll
  for (int ddy = -SPAN; ddy <= SPAN; ++ddy) {
    int yy = y + ddy;
    if (yy < 0 || yy >= HB) continue;
#pragma unroll
    for (int ddx = -SPAN; ddx <= SPAN; ++ddx) {
      int xx = x + ddx;
      if (xx < 0 || xx >= WB) continue;
      float dsp = (16.0f / 9.0f) * (float)(ddx * ddx + ddy * ddy);
      sn += expf(-0.5f * dsp);
      const float* q = base + ((size_t)yy * WB + xx) * 3;
      float d0 = q[0] - r0, d1 = q[1] - r1, d2 = q[2] - r2;
      bn += expf(-0.5f * (dsp + d0 * d0 + d1 * d1 + d2 * d2));
    }
  }
  const float EPSF = 1e-20f;
  spf[idx] = sn / (sn + EPSF);
  blf[idx] = sn / (bn + EPSF);
}

// ---------------------------------------------------------------------------
// Kernel 3: a = rowsum(compat @ ws), b = rowsum(compat @ wbk) via
// V_WMMA_F32_16X16X4_F32. 1 block x 128 threads = 4 waves; each wave owns one
// 16x16 tile of the padded 32x32 products, K looped 0..23 (21 padded to 24).
// ---------------------------------------------------------------------------
__global__ __launch_bounds__(128) void ab_wmma_kernel(
    const float* __restrict__ ws_m, const float* __restrict__ wbk_m,
    const float* __restrict__ compat_m, float* __restrict__ ab) {
  __shared__ float Acm[32][32];   // compat, zero-padded
  __shared__ float Bws[32][32];   // ws, zero-padded
  __shared__ float Bwb[32][32];   // wbk, zero-padded
  __shared__ float M1[32][32];    // compat @ ws
  __shared__ float M2[32][32];    // compat @ wbk

  int t = threadIdx.x;
  for (int i = t; i < 1024; i += 128) {
    int r = i >> 5, c = i & 31;
    bool in = (r < C) && (c < C);
    Acm[r][c] = in ? compat_m[r * C + c] : 0.f;
    Bws[r][c] = in ? ws_m[r * C + c]     : 0.f;
    Bwb[r][c] = in ? wbk_m[r * C + c]    : 0.f;
  }
  __syncthreads();

  int wave = t >> 5;
  int lane = t & 31;
  int ti = (wave >> 1) * 16;          // output tile row base
  int tj = (wave & 1) * 16;           // output tile col base
  int l15   = lane & 15;
  int khalf = (lane >> 4) * 2;        // lanes 16-31 carry K+2..K+3

  v8f d1 = {};
  v8f d2 = {};
  for (int k0 = 0; k0 < 24; k0 += 4) {
    // A fragment (16x4): lane -> row l15, K = k0+khalf (+1 in 2nd VGPR)
    v2f a;
    a[0] = Acm[ti + l15][k0 + khalf];
    a[1] = Acm[ti + l15][k0 + khalf + 1];
    // B fragment (4x16): lane -> col l15, same K mapping
    v2f b1;
    b1[0] = Bws[k0 + khalf][tj + l15];
    b1[1] = Bws[k0 + khalf + 1][tj + l15];
    v2f b2;
    b2[0] = Bwb[k0 + khalf][tj + l15];
    b2[1] = Bwb[k0 + khalf + 1][tj + l15];
    d1 = __builtin_amdgcn_wmma_f32_16x16x4_f32(false, a, false, b1,
                                               (short)0, d1, false, false);
    d2 = __builtin_amdgcn_wmma_f32_16x16x4_f32(false, a, false, b2,
                                               (short)0, d2, false, false);
  }

  // D layout: VGPR r, lanes 0-15 -> M=r, N=lane; lanes 16-31 -> M=r+8, N=lane-16
  int rbase = (lane >> 4) * 8;
#pragma unroll
  for (int r = 0; r < 8; ++r) {
    M1[ti + rbase + r][tj + l15] = d1[r];
    M2[ti + rbase + r][tj + l15] = d2[r];
  }
  __syncthreads();

  if (t < C) {
    float sa = 0.f, sb = 0.f;
    for (int j = 0; j < C; ++j) { sa += M1[t][j]; sb += M2[t][j]; }
    ab[t]      = sa;   // a[c]
    ab[32 + t] = sb;   // b[c]
  }
}

// ---------------------------------------------------------------------------
// Kernel 4: streaming pass. out(b,Y,X,c) = unary - (a[c]*SP + b[c]*BL), where
// SP/BL are legacy-bilinear upsampled scalar fields (128 -> 512: exact
// quarter-step weights).
// ---------------------------------------------------------------------------
__global__ __launch_bounds__(256) void output_kernel(
    const float* __restrict__ unary, const float* __restrict__ spf,
    const float* __restrict__ blf, const float* __restrict__ ab,
    float* __restrict__ out) {
  __shared__ float sa[32], sb[32];
  if (threadIdx.x < 32) {
    sa[threadIdx.x] = ab[threadIdx.x];
    sb[threadIdx.x] = ab[32 + threadIdx.x];
  }
  __syncthreads();

  int pix = blockIdx.x * blockDim.x + threadIdx.x;
  if (pix >= BATCH * H * W) return;
  int X  = pix % W;
  int Y  = (pix / W) % H;
  int bb = pix / (W * H);

  int y0 = Y >> 2;
  int y1 = y0 + 1; if (y1 > HB - 1) y1 = HB - 1;
  float wy = (float)(Y & 3) * 0.25f;
  int x0 = X >> 2;
  int x1 = x0 + 1; if (x1 > WB - 1) x1 = WB - 1;
  float wx = (float)(X & 3) * 0.25f;

  const float* sf = spf + (size_t)bb * HB * WB;
  const float* bf = blf + (size_t)bb * HB * WB;
  float s00 = sf[y0 * WB + x0], s01 = sf[y0 * WB + x1];
  float s10 = sf[y1 * WB + x0], s11 = sf[y1 * WB + x1];
  float b00 = bf[y0 * WB + x0], b01 = bf[y0 * WB + x1];
  float b10 = bf[y1 * WB + x0], b11 = bf[y1 * WB + x1];

  // rows = v(y0)*(1-wy) + v(y1)*wy; out = rows(x0)*(1-wx) + rows(x1)*wx
  float SPv = (s00 * (1.f - wy) + s10 * wy) * (1.f - wx)
            + (s01 * (1.f - wy) + s11 * wy) * wx;
  float BLv = (b00 * (1.f - wy) + b10 * wy) * (1.f - wx)
            + (b01 * (1.f - wy) + b11 * wy) * wx;

  size_t base = (size_t)pix * C;
#pragma unroll
  for (int c = 0; c < C; ++c) {
    out[base + c] = unary[base + c] - (sa[c] * SPv + sb[c] * BLv);
  }
}

// ---------------------------------------------------------------------------
extern "C" void kernel_launch(void* const* d_in, const int* in_sizes, int n_in,
                              void* d_out, int out_size, void* d_ws, size_t ws_size,
                              hipStream_t stream) {
  (void)in_sizes; (void)n_in; (void)out_size; (void)ws_size;
  const float* unary  = (const float*)d_in[0];
  const float* rgb    = (const float*)d_in[1];
  const float* ws_m   = (const float*)d_in[2];
  const float* wbk_m  = (const float*)d_in[3];
  const float* compat = (const float*)d_in[4];
  float* out = (float*)d_out;

  float* scratch = (float*)d_ws;
  float* rgb_ds = scratch;                                   // B*HB*WB*3
  float* spf    = rgb_ds + (size_t)BATCH * HB * WB * 3;      // B*HB*WB
  float* blf    = spf    + (size_t)BATCH * HB * WB;          // B*HB*WB
  float* ab     = blf    + (size_t)BATCH * HB * WB;          // 64

  int nlo = BATCH * HB * WB;  // 65536
  pool_rgb_kernel<<<(nlo + 255) / 256, 256, 0, stream>>>(rgb, rgb_ds);
  fields_kernel<<<(nlo + 255) / 256, 256, 0, stream>>>(rgb_ds, spf, blf);
  ab_wmma_kernel<<<1, 128, 0, stream>>>(ws_m, wbk_m, compat, ab);

  int nhi = BATCH * H * W;    // 1048576
  output_kernel<<<(nhi + 255) / 256, 256, 0, stream>>>(unary, spf, blf, ab, out);
}